// LatentDecoder_27590869909739
// MI455X (gfx1250) — compile-verified
//
#include <hip/hip_runtime.h>
#include <math.h>

#define NRES 8000
#define KNB  30
#define NEDGE (NRES*KNB)

typedef __attribute__((ext_vector_type(2))) float v2f;
typedef __attribute__((ext_vector_type(8))) float v8f;

// ---------------------------------------------------------------- utilities
__device__ __forceinline__ float actf(float v, int act){
  if (act==1) return v/(1.f+__expf(-v));      // silu
  if (act==2) return 1.f/(1.f+__expf(-v));    // sigmoid
  return v;
}

struct F3{float x,y,z;};
__device__ __forceinline__ F3 f3sub(F3 a,F3 b){return {a.x-b.x,a.y-b.y,a.z-b.z};}
__device__ __forceinline__ F3 f3add(F3 a,F3 b){return {a.x+b.x,a.y+b.y,a.z+b.z};}
__device__ __forceinline__ float f3dot(F3 a,F3 b){return a.x*b.x+a.y*b.y+a.z*b.z;}
__device__ __forceinline__ F3 f3cross(F3 a,F3 b){
  return {a.y*b.z-a.z*b.y, a.z*b.x-a.x*b.z, a.x*b.y-a.y*b.x};}
__device__ __forceinline__ F3 f3norm(F3 a){
  float n = sqrtf(f3dot(a,a)) + 1e-8f;
  return {a.x/n, a.y/n, a.z/n};
}
__device__ __forceinline__ float fixnum(float v){ return isfinite(v)?v:0.f; }

// ================================================================ kNN
__global__ void knn_kernel(const float* __restrict__ bb,
                           const unsigned char* __restrict__ xmask,
                           int* __restrict__ nbr)
{
  __shared__ float sd[NRES];
  __shared__ float rv[256];
  __shared__ int   ri[256];
  const int i = blockIdx.x;
  const int t = threadIdx.x;
  const bool mi = xmask[i]!=0;
  const float xi0 = mi?1.0e6f:bb[(size_t)i*12+3];
  const float xi1 = mi?1.0e6f:bb[(size_t)i*12+4];
  const float xi2 = mi?1.0e6f:bb[(size_t)i*12+5];
  const float x2i = xi0*xi0+xi1*xi1+xi2*xi2;
  for (int j=t;j<NRES;j+=256){
    const bool mj = xmask[j]!=0;
    float a = mj?1.0e6f:bb[(size_t)j*12+3];
    float b = mj?1.0e6f:bb[(size_t)j*12+4];
    float c = mj?1.0e6f:bb[(size_t)j*12+5];
    float d2 = x2i + (a*a+b*b+c*c) - 2.f*(xi0*a+xi1*b+xi2*c);
    if (j==i) d2 = 1.0e12f;
    sd[j]=d2;
  }
  __syncthreads();
  for (int kk=0;kk<KNB;kk++){
    float bv=3.0e38f; int bi=NRES;
    for (int j=t;j<NRES;j+=256){
      float v=sd[j];
      if (v<bv || (v==bv && j<bi)){bv=v;bi=j;}
    }
    rv[t]=bv; ri[t]=bi;
    __syncthreads();
    for (int s=128;s>0;s>>=1){
      if (t<s){
        float ov=rv[t+s]; int oi=ri[t+s];
        if (ov<rv[t] || (ov==rv[t] && oi<ri[t])){rv[t]=ov; ri[t]=oi;}
      }
      __syncthreads();
    }
    if (t==0){ nbr[(size_t)i*KNB+kk]=ri[0]; sd[ri[0]]=3.9e38f; }
    __syncthreads();
  }
}

// ================================================================ node init
__device__ __forceinline__ F3 loadAtom(const float* bb, long a){
  long r=a/3, s=a%3;
  const float* p = bb + r*12 + s*3;
  return {p[0],p[1],p[2]};
}
__global__ void node_init(const float* __restrict__ bb, float* __restrict__ bbemb, int Nn)
{
  int n = blockIdx.x*blockDim.x + threadIdx.x;
  if (n>=Nn) return;
  float cD[3], sD[3];
  for (int s=0;s<3;s++){
    long tt = 3L*n - 1 + s;
    if (tt<0 || tt > 3L*Nn-4){ cD[s]=1.f; sD[s]=0.f; continue; }
    F3 p0=loadAtom(bb,tt), p1=loadAtom(bb,tt+1), p2=loadAtom(bb,tt+2), p3=loadAtom(bb,tt+3);
    F3 u2=f3norm(f3sub(p1,p0));
    F3 u1=f3norm(f3sub(p2,p1));
    F3 u0=f3norm(f3sub(p3,p2));
    F3 n2=f3norm(f3cross(u2,u1));
    F3 n1=f3norm(f3cross(u1,u0));
    float cd = f3dot(n2,n1);
    cd = fminf(fmaxf(cd, -1.f+1e-7f), 1.f-1e-7f);
    float sg = f3dot(u2,n1);
    float sgn = (sg>0.f)?1.f:((sg<0.f)?-1.f:0.f);
    float D = sgn*acosf(cd);
    cD[s]=cosf(D); sD[s]=sinf(D);
  }
  F3 ca = {bb[(size_t)n*12+3], bb[(size_t)n*12+4], bb[(size_t)n*12+5]};
  F3 f={0,0,0}, b={0,0,0};
  if (n<Nn-1){
    F3 can={bb[(size_t)(n+1)*12+3],bb[(size_t)(n+1)*12+4],bb[(size_t)(n+1)*12+5]};
    f = f3norm(f3sub(can,ca));
  }
  if (n>0){
    F3 cap={bb[(size_t)(n-1)*12+3],bb[(size_t)(n-1)*12+4],bb[(size_t)(n-1)*12+5]};
    b = f3norm(f3sub(cap,ca));
  }
  F3 na = {bb[(size_t)n*12+0], bb[(size_t)n*12+1], bb[(size_t)n*12+2]};
  F3 cc = {bb[(size_t)n*12+6], bb[(size_t)n*12+7], bb[(size_t)n*12+8]};
  F3 c_ = f3norm(f3sub(cc,ca));
  F3 n_ = f3norm(f3sub(na,ca));
  F3 bv = f3norm(f3add(c_,n_));
  F3 pv = f3norm(f3cross(c_,n_));
  const float s13 = 0.57735026918962576f, s23 = 0.81649658092772603f;
  F3 sc = {-bv.x*s13 - pv.x*s23, -bv.y*s13 - pv.y*s23, -bv.z*s13 - pv.z*s23};
  float* o = bbemb + (size_t)n*24;
  o[0]=cD[0]; o[1]=cD[1]; o[2]=cD[2]; o[3]=sD[0]; o[4]=sD[1]; o[5]=sD[2];
  float vf[3]={fixnum(f.x),fixnum(f.y),fixnum(f.z)};
  float vb[3]={fixnum(b.x),fixnum(b.y),fixnum(b.z)};
  float vs[3]={fixnum(sc.x),fixnum(sc.y),fixnum(sc.z)};
  for (int c=0;c<3;c++){
    float* r = o + (1+c)*6;
    r[0]=vf[c]; r[1]=vb[c]; r[2]=vs[c]; r[3]=0.f; r[4]=0.f; r[5]=0.f;
  }
}

// ================================================================ edge features
__global__ void edge_feat(const float* __restrict__ bb, const int* __restrict__ nbr,
                          float* __restrict__ efeat, float* __restrict__ rbf)
{
  int e = blockIdx.x*blockDim.x + threadIdx.x;
  if (e>=NEDGE) return;
  int i = e/KNB;
  int j = nbr[e];
  float vx = bb[(size_t)i*12+3]-bb[(size_t)j*12+3];
  float vy = bb[(size_t)i*12+4]-bb[(size_t)j*12+4];
  float vz = bb[(size_t)i*12+5]-bb[(size_t)j*12+5];
  float d = sqrtf(vx*vx+vy*vy+vz*vz);
  #pragma unroll
  for (int m=0;m<16;m++){
    float z = (d - (float)m*(20.f/15.f))*(1.f/1.25f);
    float r = __expf(-z*z);
    rbf[(size_t)e*16+m]   = r;
    efeat[(size_t)e*32+m] = r;
  }
  float diff = (float)(j - i);
  #pragma unroll
  for (int f=0;f<8;f++){
    float fr = __expf(-((float)(2*f)/16.f)*9.210340371976184f); // ln 1e4
    float a = diff*fr;
    efeat[(size_t)e*32+16+f] = cosf(a);
    efeat[(size_t)e*32+24+f] = sinf(a);
  }
}

// ================================================================ generic f32 WMMA GEMM
// D = act( A(MxK) @ B(KxN) [+ C] ); one wave computes a 16x64 tile with four
// V_WMMA_F32_16X16X4_F32 accumulators. K is a compile-time constant (full
// unroll, branch-free clamped loads -> pure global_load + v_wmma stream).
template<int KD>
__global__ void __launch_bounds__(128)
gemm_wmma_f32_t(const float* __restrict__ A, int lda,
                const float* __restrict__ B, int ldb,
                const float* __restrict__ C, int ldc,
                float* __restrict__ D, int ldd,
                int M, int Nd, int act, int hasC)
{
  const int lane = threadIdx.x & 31;
  const int wv   = threadIdx.x >> 5;
  const int m0   = (blockIdx.x*4 + wv)*16;
  const int n0   = blockIdx.y*64;
  const int half = lane >> 4;
  const int l15  = lane & 15;
  const int arow = min(m0 + l15, M-1);            // clamp: stores are guarded
  const float* Ar = A + (size_t)arow*lda + 2*half;
  const float* Bp[4];
  #pragma unroll
  for (int t=0;t<4;t++){
    int nc = min(n0 + t*16 + l15, Nd-1);          // clamp: garbage cols unstored
    Bp[t] = B + (size_t)(2*half)*ldb + nc;
  }
  v8f acc[4];
  #pragma unroll
  for (int t=0;t<4;t++)
    #pragma unroll
    for (int r=0;r<8;r++) acc[t][r]=0.f;

  #pragma unroll
  for (int k0=0;k0<KD;k0+=4){
    v2f a = *(const v2f*)(Ar + k0);               // 8B-aligned (lda even)
    #pragma unroll
    for (int t=0;t<4;t++){
      v2f b;
      b.x = Bp[t][(size_t)k0*ldb];
      b.y = Bp[t][(size_t)(k0+1)*ldb];
      acc[t] = __builtin_amdgcn_wmma_f32_16x16x4_f32(false, a, false, b,
                                                     (short)0, acc[t], false, false);
    }
  }
  #pragma unroll
  for (int t=0;t<4;t++){
    int n = n0 + t*16 + l15;
    #pragma unroll
    for (int r=0;r<8;r++){
      int m = m0 + r + 8*half;
      if (m < M && n < Nd){
        float v = acc[t][r];
        if (hasC) v += C[(size_t)m*ldc + n];
        D[(size_t)m*ldd + n] = actf(v, act);
      }
    }
  }
}

// gather-fused WMMA GEMM for the edge MLP input:
// A row e = concat( efeat[e](32), l0[nbr[e]](64), l0[e/K](64) ), K=160, N=64.
// Segment boundaries (32/96) are multiples of 4 and ka = k0+2*half stays inside
// one segment per k0, so the unrolled loop resolves segment choice statically.
__global__ void __launch_bounds__(128)
gemm_wmma_edgecat(const float* __restrict__ efeat, const float* __restrict__ x, int ldx,
                  const int* __restrict__ nbr, const float* __restrict__ B,
                  float* __restrict__ D, int M, int act)
{
  const int lane = threadIdx.x & 31;
  const int wv   = threadIdx.x >> 5;
  const int m0   = (blockIdx.x*4 + wv)*16;
  const int half = lane >> 4;
  const int l15  = lane & 15;
  const int e    = m0 + l15;
  const int ec   = min(e, M-1);                   // clamp: stores guarded
  const int j    = nbr[ec];
  const int ii   = ec/KNB;
  const float* p0 = efeat + (size_t)ec*32 + 2*half;
  const float* p1 = x + (size_t)j*ldx  + 2*half;
  const float* p2 = x + (size_t)ii*ldx + 2*half;
  const float* Bp[4];
  #pragma unroll
  for (int t=0;t<4;t++) Bp[t] = B + (size_t)(2*half)*64 + t*16 + l15;

  v8f acc[4];
  #pragma unroll
  for (int t=0;t<4;t++)
    #pragma unroll
    for (int r=0;r<8;r++) acc[t][r]=0.f;

  #pragma unroll
  for (int k0=0;k0<160;k0+=4){
    const float* src = (k0<32) ? (p0 + k0) : (k0<96) ? (p1 + (k0-32)) : (p2 + (k0-96));
    v2f a = *(const v2f*)src;                     // 8B-aligned (even offsets)
    #pragma unroll
    for (int t=0;t<4;t++){
      v2f b;
      b.x = Bp[t][(size_t)k0*64];
      b.y = Bp[t][(size_t)(k0+1)*64];
      acc[t] = __builtin_amdgcn_wmma_f32_16x16x4_f32(false, a, false, b,
                                                     (short)0, acc[t], false, false);
    }
  }
  #pragma unroll
  for (int t=0;t<4;t++){
    int n = t*16 + l15;
    #pragma unroll
    for (int r=0;r<8;r++){
      int m = m0 + r + 8*half;
      if (m < M) D[(size_t)m*64 + n] = actf(acc[t][r], act);
    }
  }
}

// ================================================================ SO(3) conv 1
// (rotations cancel: R^T R = I). out = mean_k (src .* gate) @ W + x @ Wself.
__global__ void so3conv1(const float* __restrict__ bbemb, const float* __restrict__ latent,
                         const int* __restrict__ nbr, const float* __restrict__ gate,
                         const float* __restrict__ W, const float* __restrict__ Wself,
                         float* __restrict__ xA)
{
  const int i = blockIdx.x;
  const int lane = threadIdx.x;   // 32 lanes = 32 output channels
  float Wl[6], Ws[6];
  #pragma unroll
  for (int c=0;c<6;c++){ Wl[c]=W[c*32+lane]; Ws[c]=Wself[c*32+lane]; }
  float acc[4]={0.f,0.f,0.f,0.f};
  for (int k=0;k<KNB;k++){
    const int e = i*KNB + k;
    const int j = nbr[e];
    float g6[6];
    #pragma unroll
    for (int c=0;c<6;c++) g6[c]=gate[(size_t)e*6+c];
    #pragma unroll
    for (int r=0;r<4;r++){
      const float* s = bbemb + (size_t)j*24 + r*6;
      #pragma unroll
      for (int c=0;c<6;c++) acc[r] += s[c]*g6[c]*Wl[c];
    }
  }
  #pragma unroll
  for (int r=0;r<4;r++){
    const float* s = bbemb + (size_t)i*24 + r*6;
    float self=0.f;
    #pragma unroll
    for (int c=0;c<6;c++) self += s[c]*Ws[c];
    xA[(size_t)i*256 + r*64 + lane]      = acc[r]*(1.f/30.f) + self;
    xA[(size_t)i*256 + r*64 + 32 + lane] = latent[(size_t)i*128 + r*32 + lane];
  }
}

// ================================================================ attention pieces
__global__ void kq_kernel(const float* __restrict__ q, const float* __restrict__ Wk,
                          float* __restrict__ kq)
{
  const int i = blockIdx.x;
  const int c = threadIdx.x;          // 96 threads
  const float* qi = q + (size_t)i*128;
  const float* wr = Wk + (size_t)c*128;
  #pragma unroll
  for (int h=0;h<8;h++){
    float s=0.f;
    #pragma unroll
    for (int a=0;a<16;a++) s += wr[h*16+a]*qi[h*16+a];
    kq[(size_t)i*768 + c*8 + h] = s;
  }
}

__global__ void attn_kernel(const float* __restrict__ x, const float* __restrict__ efeat,
                            const int* __restrict__ nbr, const float* __restrict__ kq,
                            float* __restrict__ attn, int ldx)
{
  const int i = blockIdx.x;
  const int lane = threadIdx.x;
  const bool ok = lane < KNB;
  const int e = i*KNB + (ok?lane:0);
  const int j = ok ? nbr[e] : 0;
  const float* kqi = kq + (size_t)i*768;
  float acc[8];
  #pragma unroll
  for (int h=0;h<8;h++) acc[h]=0.f;
  for (int c=0;c<96;c++){
    float v;
    if (c<64) v = ok ? x[(size_t)j*ldx + c] : 0.f;
    else      v = ok ? efeat[(size_t)e*32 + (c-64)] : 0.f;
    const float* kc = kqi + c*8;
    #pragma unroll
    for (int h=0;h<8;h++) acc[h] += v*kc[h];
  }
  #pragma unroll
  for (int h=0;h<8;h++){
    float l = acc[h];
    l = (l>0.f ? l : 0.01f*l) * 0.25f;        // leaky_relu / sqrt(16)
    if (!ok) l = -3.4e38f;
    float m = l;
    #pragma unroll
    for (int o=16;o>0;o>>=1) m = fmaxf(m, __shfl_xor(m, o));
    float ev = ok ? __expf(l - m) : 0.f;
    float s = ev;
    #pragma unroll
    for (int o=16;o>0;o>>=1) s += __shfl_xor(s, o);
    if (ok) attn[(size_t)e*8 + h] = ev/s;
  }
}

__global__ void tkern(const float* __restrict__ x, const int* __restrict__ nbr,
                      const float* __restrict__ attn, float* __restrict__ T, int ldx)
{
  const int i = blockIdx.x;
  const int t = threadIdx.x;        // 256 = 4 rows x 64 channels
  const int row = t>>6, c = t&63;
  float acc[8];
  #pragma unroll
  for (int h=0;h<8;h++) acc[h]=0.f;
  for (int k=0;k<KNB;k++){
    const int e = i*KNB+k;
    const int j = nbr[e];
    const float xv = x[(size_t)j*ldx + row*64 + c];
    const float* a = attn + (size_t)e*8;
    #pragma unroll
    for (int h=0;h<8;h++) acc[h] += a[h]*xv;
  }
  #pragma unroll
  for (int h=0;h<8;h++)
    T[(((size_t)i*8+h)*4+row)*64 + c] = acc[h];
}

__global__ void vagg2(const float* __restrict__ T, const float* __restrict__ Wv,
                      float* __restrict__ aggpre)
{
  const int i = blockIdx.x;
  const int t = threadIdx.x;        // 256
  const int row = t>>6, co = t&63, h = co>>3;
  const float* Tr = T + (((size_t)i*8+h)*4+row)*64;
  float acc=0.f;
  for (int ci=0;ci<64;ci++) acc += Tr[ci]*Wv[(size_t)ci*64+co];
  aggpre[((size_t)i*4+row)*64+co] = acc;
}

__global__ void ff_epi(const float* __restrict__ xa, const float* __restrict__ u,
                       const float* __restrict__ g, float* __restrict__ xout,
                       int C, int Nn)
{
  size_t idx = (size_t)blockIdx.x*blockDim.x + threadIdx.x;
  size_t tot = (size_t)Nn*3*C;
  if (idx>=tot) return;
  int c = (int)(idx % C);
  size_t r0 = idx / C;
  int r = (int)(r0 % 3) + 1;
  size_t i = r0/3;
  size_t off = (i*4 + r)*C + c;
  xout[off] = xa[off] + u[off]*g[i*C + c];
}

__global__ void wp_kernel(const float* __restrict__ x32, const int* __restrict__ nbr,
                          const float* __restrict__ gate, float* __restrict__ wp)
{
  const int i = blockIdx.x;
  const int t = threadIdx.x;        // 128 = 4 x 32
  const int row = t>>5, c = t&31;
  float acc=0.f;
  for (int k=0;k<KNB;k++){
    const int e = i*KNB+k;
    const int j = nbr[e];
    acc += x32[(size_t)j*128 + row*32 + c]*gate[(size_t)e*32+c];
  }
  wp[((size_t)i*4+row)*32+c] = acc*(1.f/30.f);
}

// ================================================================ head
__global__ void head_kernel(const float* __restrict__ atom, const float* __restrict__ ln_g,
                            const float* __restrict__ ln_b, const float* __restrict__ seqW,
                            const float* __restrict__ seqb, float* __restrict__ out, int Nn)
{
  __shared__ float inv[91];
  __shared__ float lg[20];
  __shared__ float st[2];
  const int i = blockIdx.x;
  const int t = threadIdx.x;        // 128
  if (t<91) inv[t] = atom[((size_t)i*4)*91 + t];
  __syncthreads();
  if (t==0){
    float m=0.f;
    for (int c=0;c<91;c++) m += inv[c];
    m *= (1.f/91.f);
    float v=0.f;
    for (int c=0;c<91;c++){ float d=inv[c]-m; v+=d*d; }
    v *= (1.f/91.f);
    st[0]=m; st[1]=1.f/sqrtf(v+1e-5f);
  }
  __syncthreads();
  if (t<91) inv[t] = (inv[t]-st[0])*st[1]*ln_g[t]+ln_b[t];
  __syncthreads();
  if (t<20){
    float s=seqb[t];
    for (int c=0;c<91;c++) s += inv[c]*seqW[(size_t)c*20+t];
    lg[t]=s;
  }
  __syncthreads();
  if (t==0){
    float m=lg[0];
    for (int j=1;j<20;j++) m=fmaxf(m,lg[j]);
    float s=0.f;
    for (int j=0;j<20;j++) s+=__expf(lg[j]-m);
    st[0]=m; st[1]=logf(s);
  }
  __syncthreads();
  if (t<20) out[(size_t)Nn*273 + (size_t)i*20 + t] = lg[t]-st[0]-st[1];
  for (int tt=t; tt<273; tt+=128){
    int c = tt/3, r = tt%3;
    out[(size_t)i*273 + tt] = atom[((size_t)i*4+1+r)*91 + c];
  }
}

// ================================================================ host side
static void launch_gemm(hipStream_t s, const float* A,int lda,const float* B,int ldb,
                        const float* C,int ldc,float* D,int ldd,int M,int Kd,int Nd,int act){
  dim3 g((M+63)/64,(Nd+63)/64);
  switch (Kd){
    case 16:
      gemm_wmma_f32_t<16><<<g, dim3(128), 0, s>>>(A,lda,B,ldb,C,ldc,D,ldd,M,Nd,act,C?1:0);
      break;
    case 32:
      gemm_wmma_f32_t<32><<<g, dim3(128), 0, s>>>(A,lda,B,ldb,C,ldc,D,ldd,M,Nd,act,C?1:0);
      break;
    default:
      gemm_wmma_f32_t<64><<<g, dim3(128), 0, s>>>(A,lda,B,ldb,C,ldc,D,ldd,M,Nd,act,C?1:0);
      break;
  }
}

extern "C" void kernel_launch(void* const* d_in, const int* in_sizes, int n_in,
                              void* d_out, int out_size, void* d_ws, size_t ws_size,
                              hipStream_t stream)
{
  (void)in_sizes; (void)n_in; (void)out_size; (void)ws_size;
  const float* bb      = (const float*)d_in[0];
  const float* latent  = (const float*)d_in[1];
  const unsigned char* xmask = (const unsigned char*)d_in[2];
  const float* bb_gw1  = (const float*)d_in[3];
  const float* bb_gw2  = (const float*)d_in[4];
  const float* bb_W    = (const float*)d_in[5];
  const float* bb_Wself= (const float*)d_in[6];
  const float* Wq      = (const float*)d_in[7];
  const float* Wk      = (const float*)d_in[8];
  const float* Wv      = (const float*)d_in[9];
  const float* Wo      = (const float*)d_in[10];
  const float* fw1     = (const float*)d_in[11];
  const float* fw2     = (const float*)d_in[12];
  const float* fg      = (const float*)d_in[13];
  const float* fl1     = (const float*)d_in[14];
  const float* ew1     = (const float*)d_in[15];
  const float* ew2     = (const float*)d_in[16];
  const float* pWq     = (const float*)d_in[17];
  const float* pWk     = (const float*)d_in[18];
  const float* pWv     = (const float*)d_in[19];
  const float* pWo     = (const float*)d_in[20];
  const float* pfw1    = (const float*)d_in[21];
  const float* pfw2    = (const float*)d_in[22];
  const float* pfg     = (const float*)d_in[23];
  const float* pfl1    = (const float*)d_in[24];
  const float* a_gw1   = (const float*)d_in[25];
  const float* a_gw2   = (const float*)d_in[26];
  const float* a_W     = (const float*)d_in[27];
  const float* a_Wself = (const float*)d_in[28];
  const float* ln_g    = (const float*)d_in[29];
  const float* ln_b    = (const float*)d_in[30];
  const float* seqW    = (const float*)d_in[31];
  const float* seqb    = (const float*)d_in[32];
  float* out = (float*)d_out;

  char* wsb = (char*)d_ws;
  size_t cur = 0;
  auto carve = [&](size_t bytes)->void*{
    void* p = wsb + cur;
    cur = (cur + bytes + 255) & ~(size_t)255;
    return p;
  };
  int*   nbr    = (int*)  carve((size_t)NEDGE*4);
  float* bbemb  = (float*)carve((size_t)NRES*24*4);
  float* efeat  = (float*)carve((size_t)NEDGE*32*4);
  float* rbf    = (float*)carve((size_t)NEDGE*16*4);
  float* xA     = (float*)carve((size_t)NRES*256*4);
  float* xB     = (float*)carve((size_t)NRES*256*4);
  float* gateb  = (float*)carve((size_t)NEDGE*32*4);
  float* qb     = (float*)carve((size_t)NRES*128*4);
  float* kqb    = (float*)carve((size_t)NRES*768*4);
  float* attnb  = (float*)carve((size_t)NEDGE*8*4);
  float* aggpre = (float*)carve((size_t)NRES*256*4);
  float* ub     = (float*)carve((size_t)NRES*256*4);
  float* gb     = (float*)carve((size_t)NRES*64*4);
  float* h32    = (float*)carve((size_t)NRES*32*4);
  float* atomb  = (float*)carve((size_t)NRES*4*91*4);
  float* bigb   = (float*)carve((size_t)NRES*2048*4);   // reused: gate-hid / T / edge-hid / rbf-hid

  // ---- geometry
  knn_kernel<<<NRES,256,0,stream>>>(bb, xmask, nbr);
  node_init<<<(NRES+255)/256,256,0,stream>>>(bb, bbemb, NRES);
  edge_feat<<<(NEDGE+255)/256,256,0,stream>>>(bb, nbr, efeat, rbf);

  // ---- SO(3) conv 1  (gate MLP via WMMA, aggregate fused per node)
  launch_gemm(stream, efeat,32, bb_gw1,64, nullptr,0, bigb,64, NEDGE,32,64, 1);
  launch_gemm(stream, bigb,64, bb_gw2,6,  nullptr,0, gateb,6, NEDGE,64,6, 0);
  so3conv1<<<NRES,32,0,stream>>>(bbemb, latent, nbr, gateb, bb_W, bb_Wself, xA);

  // ---- 4 transformer layers (x lives in xA, (N,4,64))
  for (int l=0;l<4;l++){
    const float* Wq_l = Wq + (size_t)l*64*128;
    const float* Wk_l = Wk + (size_t)l*96*128;
    const float* Wv_l = Wv + (size_t)l*64*64;
    const float* Wo_l = Wo + (size_t)l*64*64;
    const float* fw1_l= fw1+ (size_t)l*64*32;
    const float* fw2_l= fw2+ (size_t)l*32*64;
    const float* fg_l = fg + (size_t)l*64*64;
    const float* fl1_l= fl1+ (size_t)l*64*64;
    const float* ew1_l= ew1+ (size_t)l*160*64;
    const float* ew2_l= ew2+ (size_t)l*64*32;

    launch_gemm(stream, xA,256, Wq_l,128, nullptr,0, qb,128, NRES,64,128, 0);
    kq_kernel<<<NRES,96,0,stream>>>(qb, Wk_l, kqb);
    attn_kernel<<<NRES,32,0,stream>>>(xA, efeat, nbr, kqb, attnb, 256);
    tkern<<<NRES,256,0,stream>>>(xA, nbr, attnb, bigb, 256);
    vagg2<<<NRES,256,0,stream>>>(bigb, Wv_l, aggpre);
    launch_gemm(stream, aggpre,64, Wo_l,64, xA,64, xB,64, 4*NRES,64,64, 0);   // xa = x + agg@Wo
    launch_gemm(stream, xB,256, fw1_l,32, nullptr,0, h32,32, NRES,64,32, 1);  // silu
    launch_gemm(stream, xB,256, fg_l,64,  nullptr,0, gb,64,  NRES,64,64, 2);  // sigmoid
    launch_gemm(stream, xB,64,  fl1_l,64, nullptr,0, ub,64,  4*NRES,64,64, 0);
    launch_gemm(stream, h32,32, fw2_l,64, xB,256, xA,256, NRES,32,64, 0);     // row0 out
    {
      size_t tot=(size_t)NRES*3*64;
      ff_epi<<<(tot+255)/256,256,0,stream>>>(xB, ub, gb, xA, 64, NRES);
    }
    // edge feature update (gather-fused WMMA GEMM)
    gemm_wmma_edgecat<<<dim3((NEDGE+63)/64),dim3(128),0,stream>>>(efeat, xA, 256, nbr,
                                                                  ew1_l, bigb, NEDGE, 1);
    launch_gemm(stream, bigb,64, ew2_l,32, nullptr,0, efeat,32, NEDGE,64,32, 0);
  }

  // ---- projection block (Wo -> 32ch, no residual)
  launch_gemm(stream, xA,256, pWq,128, nullptr,0, qb,128, NRES,64,128, 0);
  kq_kernel<<<NRES,96,0,stream>>>(qb, pWk, kqb);
  attn_kernel<<<NRES,32,0,stream>>>(xA, efeat, nbr, kqb, attnb, 256);
  tkern<<<NRES,256,0,stream>>>(xA, nbr, attnb, bigb, 256);
  vagg2<<<NRES,256,0,stream>>>(bigb, pWv, aggpre);
  launch_gemm(stream, aggpre,64, pWo,32, nullptr,0, xB,32, 4*NRES,64,32, 0);  // xa = agg
  launch_gemm(stream, xB,128, pfw1,32, nullptr,0, h32,32, NRES,32,32, 1);
  launch_gemm(stream, xB,128, pfg,32,  nullptr,0, gb,32,  NRES,32,32, 2);
  launch_gemm(stream, xB,32,  pfl1,32, nullptr,0, ub,32,  4*NRES,32,32, 0);
  launch_gemm(stream, h32,32, pfw2,32, xB,128, xA,128, NRES,32,32, 0);
  {
    size_t tot=(size_t)NRES*3*32;
    ff_epi<<<(tot+255)/256,256,0,stream>>>(xB, ub, gb, xA, 32, NRES);
  }

  // ---- atom conv (gate from rbf) + output head
  launch_gemm(stream, rbf,16, a_gw1,64, nullptr,0, bigb,64, NEDGE,16,64, 1);
  launch_gemm(stream, bigb,64, a_gw2,32, nullptr,0, gateb,32, NEDGE,64,32, 0);
  wp_kernel<<<NRES,128,0,stream>>>(xA, nbr, gateb, aggpre);
  launch_gemm(stream, aggpre,32, a_W,91,     nullptr,0, atomb,91, 4*NRES,32,91, 0);
  launch_gemm(stream, xA,32,     a_Wself,91, atomb,91,  atomb,91, 4*NRES,32,91, 0);
  head_kernel<<<NRES,128,0,stream>>>(atomb, ln_g, ln_b, seqW, seqb, out, NRES);
}